// PicoTransformerAdder_1236950582216
// MI455X (gfx1250) — compile-verified
//
#include <hip/hip_runtime.h>
#include <hip/hip_bf16.h>

typedef float v2f __attribute__((ext_vector_type(2)));
typedef float v8f __attribute__((ext_vector_type(8)));

#define S_LEN 256

// One workgroup (8 wave32) per batch element.
// Scores^T tiles via V_WMMA_F32_16X16X4_F32 (native 16x4 * 4x16 f32 shape).
// Scores carry log2(e) folded into the Q-side scale, so softmax uses raw
// v_exp_f32 (exp2) with no per-element multiply. V is rank-1 in x
// (V[k] = x_k*av + cv), so attention output needs only one weighted
// accumulator per query: out[q] = (sum_k p*x_k)*av + cv.
__global__ __launch_bounds__(256) void pico_attn_kernel(
    const float* __restrict__ x,
    const float* __restrict__ emb_w, const float* __restrict__ emb_b,
    const float* __restrict__ q_w,   const float* __restrict__ q_b,
    const float* __restrict__ k_w,   const float* __restrict__ k_b,
    const float* __restrict__ v_w,   const float* __restrict__ v_b,
    const float* __restrict__ f1_w,  const float* __restrict__ f1_b,
    const float* __restrict__ f2_w,  const float* __restrict__ f2_b,
    float* __restrict__ out)
{
    __shared__ float xs[S_LEN];
    __shared__ float pooledM;   // sum over queries of m_q = E_p[x_k]

    const int b    = blockIdx.x;
    const int tid  = threadIdx.x;
    const int lane = tid & 31;
    const int wave = tid >> 5;
    const int ln   = lane & 15;   // row/col within 16-group
    const int lh   = lane >> 4;   // half-wave: K-dim pair (0,1) vs (2,3)

    // Stage this batch's x row into LDS (1 KB).
    xs[tid] = x[b * S_LEN + tid];
    if (tid == 0) pooledM = 0.0f;
    __syncthreads();

    // Fold emb-linear + QKV projections into affine-in-x 4-vectors:
    //   Q[s] = x_s*aq + cq  (0.5/sqrt-d AND log2(e) folded into Q side),
    //   K[s] = x_s*ak + ck,  V[s] = x_s*av + cv.
    const float QSCL = 0.5f * 1.4426950408889634f;  // (1/sqrt(d)) * log2(e)
    float aq[4], cq[4], ak[4], ck[4], av[4], cv[4];
    #pragma unroll
    for (int e = 0; e < 4; ++e) {
        float sq = 0.f, sqb = 0.f, sk = 0.f, skb = 0.f, sv = 0.f, svb = 0.f;
        #pragma unroll
        for (int d = 0; d < 4; ++d) {
            const float ew = emb_w[d], eb = emb_b[d];
            sq += q_w[e*4+d]*ew;  sqb += q_w[e*4+d]*eb;
            sk += k_w[e*4+d]*ew;  skb += k_w[e*4+d]*eb;
            sv += v_w[e*4+d]*ew;  svb += v_w[e*4+d]*eb;
        }
        aq[e] = QSCL*sq; cq[e] = QSCL*(sqb + q_b[e]);
        ak[e] = sk;      ck[e] = skb + k_b[e];
        av[e] = sv;      cv[e] = svb + v_b[e];
    }

    float po = 0.f;  // per-lane partial of sum_q m_q (lh==0 lanes only)

    for (int qq = 0; qq < 2; ++qq) {
        const int qb = wave * 2 + qq;

        // B fragment (Q^T, 4x16 f32): lane holds column q = qb*16+ln;
        // V0/V1 carry K-dims (0,1) for lanes 0-15 and (2,3) for lanes 16-31.
        const float xq = xs[qb*16 + ln];
        v2f bf;
        bf.x = xq*aq[2*lh+0] + cq[2*lh+0];
        bf.y = xq*aq[2*lh+1] + cq[2*lh+1];

        float m = -3.0e38f, sden = 0.f, onum = 0.f;

        for (int kt = 0; kt < 16; ++kt) {
            // A fragment (K rows, 16x4 f32): lane holds row k = kt*16+ln.
            const float xk = xs[kt*16 + ln];
            v2f af;
            af.x = xk*ak[2*lh+0] + ck[2*lh+0];
            af.y = xk*ak[2*lh+1] + ck[2*lh+1];

            v8f c = {};
            // D[j] = log2-domain score^T for k = kt*16 + lh*8 + j, q = qb*16+ln
            v8f dacc = __builtin_amdgcn_wmma_f32_16x16x4_f32(
                false, af, false, bf, (short)0, c, false, false);

            // Online softmax (base-2) over this lane's 8 key entries.
            float tmax = fmaxf(fmaxf(fmaxf(dacc[0], dacc[1]), fmaxf(dacc[2], dacc[3])),
                               fmaxf(fmaxf(dacc[4], dacc[5]), fmaxf(dacc[6], dacc[7])));
            const float mn = fmaxf(m, tmax);
            const float cs = __builtin_amdgcn_exp2f(m - mn);
            sden *= cs; onum *= cs;
            #pragma unroll
            for (int j = 0; j < 8; ++j) {
                const float e  = __builtin_amdgcn_exp2f(dacc[j] - mn);
                const float xv = xs[kt*16 + lh*8 + j];
                sden += e;
                onum = fmaf(e, xv, onum);
            }
            m = mn;
        }

        // Merge the two half-wave key partitions (k 0..7 vs 8..15 of each tile).
        const float m2 = __shfl_xor(m, 16, 32);
        const float s2 = __shfl_xor(sden, 16, 32);
        const float n2 = __shfl_xor(onum, 16, 32);
        const float mm = fmaxf(m, m2);
        const float c1 = __builtin_amdgcn_exp2f(m - mm);
        const float c2 = __builtin_amdgcn_exp2f(m2 - mm);
        sden = sden*c1 + s2*c2;
        onum = onum*c1 + n2*c2;

        // m_q = softmax-weighted mean of x over keys; count each q once.
        const float wsel = (lh == 0) ? (1.0f / sden) : 0.0f;
        po += onum * wsel;
    }

    // Reduce sum_q m_q across the wave (lanes 16-31 carry zeros).
    #pragma unroll
    for (int off = 1; off < 32; off <<= 1)
        po += __shfl_xor(po, off, 32);
    if (lane == 0) atomicAdd(&pooledM, po);
    __syncthreads();

    // pooled[d] = (mean_q m_q)*av[d] + cv[d]; then 4 -> 8 -> 1 MLP head.
    if (tid == 0) {
        const float M = pooledM * (1.0f / 256.0f);
        float p[4];
        #pragma unroll
        for (int d = 0; d < 4; ++d) p[d] = M * av[d] + cv[d];
        float y = f2_b[0];
        #pragma unroll
        for (int j = 0; j < 8; ++j) {
            float hsum = f1_b[j];
            #pragma unroll
            for (int d = 0; d < 4; ++d) hsum += p[d] * f1_w[j*4+d];
            y += fmaxf(hsum, 0.0f) * f2_w[j];
        }
        out[b] = y;
    }
}

extern "C" void kernel_launch(void* const* d_in, const int* in_sizes, int n_in,
                              void* d_out, int out_size, void* d_ws, size_t ws_size,
                              hipStream_t stream) {
    const float* x     = (const float*)d_in[0];
    const float* emb_w = (const float*)d_in[1];
    const float* emb_b = (const float*)d_in[2];
    const float* q_w   = (const float*)d_in[3];
    const float* q_b   = (const float*)d_in[4];
    const float* k_w   = (const float*)d_in[5];
    const float* k_b   = (const float*)d_in[6];
    const float* v_w   = (const float*)d_in[7];
    const float* v_b   = (const float*)d_in[8];
    const float* f1_w  = (const float*)d_in[9];
    const float* f1_b  = (const float*)d_in[10];
    const float* f2_w  = (const float*)d_in[11];
    const float* f2_b  = (const float*)d_in[12];

    const int B = in_sizes[0] / S_LEN;   // 1024
    pico_attn_kernel<<<B, 256, 0, stream>>>(
        x, emb_w, emb_b, q_w, q_b, k_w, k_b, v_w, v_b,
        f1_w, f1_b, f2_w, f2_b, (float*)d_out);
}